// Bin_Mean_Shift_34067680591989
// MI455X (gfx1250) — compile-verified
//
#include <hip/hip_runtime.h>
#include <math.h>

#define BATCH 8
#define HW 49152          // 192*256
#define NS 3000
#define NSEED 100
#define NC 100
#define BW_R 0.25f

typedef float v2f __attribute__((ext_vector_type(2)));
typedef float v8f __attribute__((ext_vector_type(8)));

// ---- output offsets (floats, concatenated in reference return order) ----
__device__ __host__ constexpr size_t OUT_SEG   = 0;                                   // B*HW*100
__device__ __host__ constexpr size_t OUT_SSEG  = (size_t)BATCH*HW*NC;                 // 39321600
__device__ __host__ constexpr size_t OUT_SPAR  = OUT_SSEG + (size_t)BATCH*NS*NC;      // 41721600
__device__ __host__ constexpr size_t OUT_CTR   = OUT_SPAR + (size_t)BATCH*3*NS;       // 41793600
__device__ __host__ constexpr size_t OUT_CMASK = OUT_CTR + (size_t)BATCH*NC*2;        // 41795200
__device__ __host__ constexpr size_t OUT_SPROB = OUT_CMASK + (size_t)BATCH*NC;        // 41796000
__device__ __host__ constexpr size_t OUT_SSEGT = OUT_SPROB + (size_t)BATCH*NS;        // 41820000

// ---- workspace offsets (floats) ----
__device__ __host__ constexpr size_t WS_SX   = 0;
__device__ __host__ constexpr size_t WS_SY   = WS_SX + (size_t)BATCH*NS;
__device__ __host__ constexpr size_t WS_SP   = WS_SY + (size_t)BATCH*NS;
__device__ __host__ constexpr size_t WS_SEED = WS_SP + (size_t)BATCH*NS;
__device__ __host__ constexpr size_t WS_MASK = WS_SEED + (size_t)BATCH*NSEED*2;
__device__ __host__ constexpr size_t WS_CTR  = WS_MASK + (size_t)BATCH*NSEED;
__device__ __host__ constexpr size_t WS_CM   = WS_CTR + (size_t)BATCH*NC*2;

// -------------------------------------------------------------------------
// Kernel 1: gather sampled data, write small outputs, min/max -> seed grid
// -------------------------------------------------------------------------
__global__ __launch_bounds__(256) void gather_kernel(
    const float* __restrict__ logit, const float* __restrict__ emb,
    const float* __restrict__ par, const int* __restrict__ seg,
    const int* __restrict__ ridx, float* __restrict__ out, float* __restrict__ ws) {
  const int b = blockIdx.x;
  const int tid = threadIdx.x;
  __shared__ float mnx[256], mny[256], mxx[256], mxy[256];
  __shared__ float se[4];

  const int* rb = ridx + (size_t)b*NS;
  const float* ex = emb + (size_t)b*2*HW;
  const float* ey = ex + HW;
  const float* lg = logit + (size_t)b*HW;
  const float* p0 = par + (size_t)b*3*HW;
  const int* sg = seg + (size_t)b*HW;

  float lmnx = 3.4e38f, lmny = 3.4e38f, lmxx = -3.4e38f, lmxy = -3.4e38f;
  for (int i = tid; i < NS; i += 256) {
    int idx = rb[i];
    float x = ex[idx], y = ey[idx];
    ws[WS_SX + (size_t)b*NS + i] = x;
    ws[WS_SY + (size_t)b*NS + i] = y;
    float pr = 1.0f / (1.0f + __expf(-lg[idx]));
    ws[WS_SP + (size_t)b*NS + i] = pr;
    out[OUT_SPROB + (size_t)b*NS + i] = pr;
    out[OUT_SSEGT + (size_t)b*NS + i] = (float)sg[idx];
    out[OUT_SPAR + (size_t)b*3*NS + 0*NS + i] = p0[idx];
    out[OUT_SPAR + (size_t)b*3*NS + 1*NS + i] = p0[HW + idx];
    out[OUT_SPAR + (size_t)b*3*NS + 2*NS + i] = p0[2*HW + idx];
    lmnx = fminf(lmnx, x); lmny = fminf(lmny, y);
    lmxx = fmaxf(lmxx, x); lmxy = fmaxf(lmxy, y);
  }
  mnx[tid] = lmnx; mny[tid] = lmny; mxx[tid] = lmxx; mxy[tid] = lmxy;
  __syncthreads();
  for (int s2 = 128; s2 > 0; s2 >>= 1) {
    if (tid < s2) {
      mnx[tid] = fminf(mnx[tid], mnx[tid + s2]);
      mny[tid] = fminf(mny[tid], mny[tid + s2]);
      mxx[tid] = fmaxf(mxx[tid], mxx[tid + s2]);
      mxy[tid] = fmaxf(mxy[tid], mxy[tid + s2]);
    }
    __syncthreads();
  }
  if (tid == 0) {
    float a = mnx[0], bb = mny[0], c = mxx[0], d = mxy[0];
    se[0] = a + (c - a) / 22.0f;   // start x
    se[1] = bb + (d - bb) / 22.0f; // start y
    se[2] = c - (c - a) / 22.0f;   // end x
    se[3] = d - (d - bb) / 22.0f;  // end y
  }
  __syncthreads();
  if (tid < NSEED) {
    int i = tid / 10, j = tid % 10;
    float xs = se[0] + (se[2] - se[0]) * (float)i / 9.0f;
    float ys = se[1] + (se[3] - se[1]) * (float)j / 9.0f;
    float* sp = ws + WS_SEED + ((size_t)b*NSEED + tid)*2;
    sp[0] = xs; sp[1] = ys;
  }
}

// -------------------------------------------------------------------------
// Kernel 2: mean shift — one wave per (seed, batch); 5 iters + 2 counts
// -------------------------------------------------------------------------
__device__ __forceinline__ float wave_sum32(float v) {
  v += __shfl_xor(v, 1, 32);
  v += __shfl_xor(v, 2, 32);
  v += __shfl_xor(v, 4, 32);
  v += __shfl_xor(v, 8, 32);
  v += __shfl_xor(v, 16, 32);
  return v;
}

__global__ __launch_bounds__(32) void meanshift_kernel(float* __restrict__ ws) {
  const int s = blockIdx.x;
  const int b = blockIdx.y;
  const int lane = threadIdx.x;
  const float* px = ws + WS_SX + (size_t)b*NS;
  const float* py = ws + WS_SY + (size_t)b*NS;
  const float* pp = ws + WS_SP + (size_t)b*NS;
  float* seedp = ws + WS_SEED + ((size_t)b*NSEED + s)*2;
  float sx = seedp[0], sy = seedp[1];

  // count at initial position
  float c0 = 0.0f;
  for (int j = lane; j < NS; j += 32) {
    float dx = sx - px[j], dy = sy - py[j];
    float d = sqrtf(dx*dx + dy*dy + 1e-12f);
    if (d < BW_R) c0 += pp[j];
  }
  c0 = wave_sum32(c0);

  for (int it = 0; it < 5; it++) {
    float sw = 0.0f, swx = 0.0f, swy = 0.0f;
    for (int j = lane; j < NS; j += 32) {
      float dx = sx - px[j], dy = sy - py[j];
      float d2 = dx*dx + dy*dy;
      float k = __expf(-8.0f * (d2 + 1e-12f));  // -0.5/BW^2 = -8
      float w = k * pp[j];
      sw += w; swx += w * px[j]; swy += w * py[j];
    }
    sw = wave_sum32(sw); swx = wave_sum32(swx); swy = wave_sum32(swy);
    sx = swx / sw; sy = swy / sw;
  }

  float c1 = 0.0f;
  for (int j = lane; j < NS; j += 32) {
    float dx = sx - px[j], dy = sy - py[j];
    float d = sqrtf(dx*dx + dy*dy + 1e-12f);
    if (d < BW_R) c1 += pp[j];
  }
  c1 = wave_sum32(c1);

  if (lane == 0) {
    seedp[0] = sx; seedp[1] = sy;
    ws[WS_MASK + (size_t)b*NSEED + s] = (c0 > 3.0f && c1 > 10.0f) ? 1.0f : 0.0f;
  }
}

// -------------------------------------------------------------------------
// Kernel 3: NMS (stable argsort + greedy) and cluster centers
// -------------------------------------------------------------------------
__global__ __launch_bounds__(128) void nms_kernel(float* __restrict__ ws, float* __restrict__ out) {
  const int b = blockIdx.x;
  const int tid = threadIdx.x;
  __shared__ float sx[NSEED], sy[NSEED], inten[NSEED];
  __shared__ int smask[NSEED], order[NSEED], labels[NSEED], alive[NSEED];

  if (tid < NSEED) {
    sx[tid] = ws[WS_SEED + ((size_t)b*NSEED + tid)*2];
    sy[tid] = ws[WS_SEED + ((size_t)b*NSEED + tid)*2 + 1];
    smask[tid] = ws[WS_MASK + (size_t)b*NSEED + tid] > 0.5f ? 1 : 0;
  }
  __syncthreads();
  if (tid < NSEED) {
    float cnt = -1.0f;
    if (smask[tid]) {
      int c = 0;
      for (int j = 0; j < NSEED; j++) {
        if (smask[j]) {
          float dx = sx[tid]-sx[j], dy = sy[tid]-sy[j];
          if (sqrtf(dx*dx + dy*dy + 1e-12f) < BW_R) c++;
        }
      }
      cnt = (float)c;
    }
    inten[tid] = cnt;
  }
  __syncthreads();
  if (tid == 0) {
    float tmp[NSEED];
    for (int i = 0; i < NSEED; i++) tmp[i] = inten[i];
    // stable descending argsort (first-max selection == jnp.argsort(-x) stable)
    for (int k = 0; k < NSEED; k++) {
      int bi = 0; float bv = tmp[0];
      for (int j = 1; j < NSEED; j++) { if (tmp[j] > bv) { bv = tmp[j]; bi = j; } }
      order[k] = bi; tmp[bi] = -1e30f;
    }
    for (int i = 0; i < NSEED; i++) labels[i] = -1;
    for (int i = 0; i < NSEED; i++) alive[i] = smask[order[i]];
    int cur = 0;
    for (int i = 0; i < NSEED; i++) {
      int ii = order[i];
      if (alive[i]) {
        labels[ii] = cur;
        for (int j = i + 1; j < NSEED; j++) {
          if (alive[j]) {
            int jj = order[j];
            float dx = sx[ii]-sx[jj], dy = sy[ii]-sy[jj];
            if (sqrtf(dx*dx + dy*dy + 1e-12f) < BW_R) { labels[jj] = cur; alive[j] = 0; }
          }
        }
        cur++;
      }
    }
  }
  __syncthreads();
  if (tid < NC) {
    int col = 0; float sxx = 0.0f, syy = 0.0f;
    for (int j = 0; j < NSEED; j++) {
      if (labels[j] == tid) { col++; sxx += sx[j]; syy += sy[j]; }
    }
    float cx = col ? sxx / (float)col : 0.0f;
    float cy = col ? syy / (float)col : 0.0f;
    float cmv = col > 0 ? 1.0f : 0.0f;
    ws[WS_CTR + ((size_t)b*NC + tid)*2]     = cx;
    ws[WS_CTR + ((size_t)b*NC + tid)*2 + 1] = cy;
    ws[WS_CM + (size_t)b*NC + tid]          = cmv;
    out[OUT_CTR + ((size_t)b*NC + tid)*2]     = cx;
    out[OUT_CTR + ((size_t)b*NC + tid)*2 + 1] = cy;
    out[OUT_CMASK + (size_t)b*NC + tid]       = cmv;
  }
}

// -------------------------------------------------------------------------
// Kernel 4: segmentation = softmax(1/(dist+0.01)) via WMMA f32 16x16x4
//   d^2 tile (16 px x 16 centers) = A x B with
//   A[m] = [x, y, x^2+y^2, 1],  B[:,n] = [-2cx, -2cy, 1, cx^2+cy^2]
// Hot path uses single-instruction v_sqrt_f32 / v_rcp_f32 / v_exp_f32 so the
// lane-divergent center mask lowers to v_cndmask instead of exec branches.
// -------------------------------------------------------------------------
__global__ __launch_bounds__(256) void seg_kernel(
    const float* __restrict__ emb, const float* __restrict__ ws, float* __restrict__ out) {
  const int b = blockIdx.y;
  const int lane = threadIdx.x & 31;
  const int nhalf = lane & 15;
  const bool lowhalf = lane < 16;
  const int wave = blockIdx.x * (blockDim.x >> 5) + (threadIdx.x >> 5);
  const int nwaves = gridDim.x * (blockDim.x >> 5);

  const float* ctr = ws + WS_CTR + (size_t)b*NC*2;
  const float* cmk = ws + WS_CM + (size_t)b*NC;
  float cxv[7], cyv[7]; bool cval[7];
#pragma unroll
  for (int t = 0; t < 7; t++) {
    int c = t*16 + nhalf;
    if (c < NC) { cxv[t] = ctr[2*c]; cyv[t] = ctr[2*c+1]; cval[t] = cmk[c] > 0.5f; }
    else        { cxv[t] = 0.0f;     cyv[t] = 0.0f;       cval[t] = false; }
  }
  v2f Bm[7];
#pragma unroll
  for (int t = 0; t < 7; t++) {
    Bm[t].x = lowhalf ? (-2.0f*cxv[t]) : 1.0f;                          // rows K=0 / K=2
    Bm[t].y = lowhalf ? (-2.0f*cyv[t]) : (cxv[t]*cxv[t]+cyv[t]*cyv[t]); // rows K=1 / K=3
  }

  const float* ex = emb + (size_t)b*2*HW;
  const float* ey = ex + HW;
  float* segb = out + OUT_SEG + (size_t)b*HW*NC;

  for (int tile = wave; tile < HW/16; tile += nwaves) {
    int p = tile*16 + nhalf;
    float x = ex[p], y = ey[p];
    v2f Am;
    Am.x = lowhalf ? x : (x*x + y*y);  // cols K=0 / K=2
    Am.y = lowhalf ? y : 1.0f;         // cols K=1 / K=3
    v8f acc[7];
#pragma unroll
    for (int t = 0; t < 7; t++) {
      v8f cz = {0.f,0.f,0.f,0.f,0.f,0.f,0.f,0.f};
      acc[t] = __builtin_amdgcn_wmma_f32_16x16x4_f32(
          false, Am, false, Bm[t], (short)0, cz, false, false);
    }
    int rowbase = tile*16 + (lowhalf ? 0 : 8);
#pragma unroll
    for (int r = 0; r < 8; r++) {
      float sv[7];
      float m = -3.4e38f;
#pragma unroll
      for (int t = 0; t < 7; t++) {
        float d2 = fmaxf(acc[t][r], 0.0f);
        float d  = __builtin_amdgcn_sqrtf(d2 + 1e-12f);   // v_sqrt_f32
        float s  = __builtin_amdgcn_rcpf(d + 0.01f);      // v_rcp_f32
        s = cval[t] ? s : -3.4e38f;                       // v_cndmask
        sv[t] = s; m = fmaxf(m, s);
      }
      m = fmaxf(m, __shfl_xor(m, 1, 16));
      m = fmaxf(m, __shfl_xor(m, 2, 16));
      m = fmaxf(m, __shfl_xor(m, 4, 16));
      m = fmaxf(m, __shfl_xor(m, 8, 16));
      float ssum = 0.0f;
#pragma unroll
      for (int t = 0; t < 7; t++) {
        // masked lanes: sv - m = -3.4e38 - m -> exp underflows to exactly 0
        float e = __expf(sv[t] - m);
        sv[t] = e; ssum += e;
      }
      ssum += __shfl_xor(ssum, 1, 16);
      ssum += __shfl_xor(ssum, 2, 16);
      ssum += __shfl_xor(ssum, 4, 16);
      ssum += __shfl_xor(ssum, 8, 16);
      float inv = __builtin_amdgcn_rcpf(ssum);            // v_rcp_f32
      float* rowp = segb + (size_t)(rowbase + r) * NC;
#pragma unroll
      for (int t = 0; t < 7; t++) {
        int n = t*16 + nhalf;
        if (n < NC) rowp[n] = sv[t] * inv;
      }
    }
  }
}

// -------------------------------------------------------------------------
// Kernel 5: sampled_seg = segmentation[ridx]
// -------------------------------------------------------------------------
__global__ __launch_bounds__(256) void sample_kernel(
    const int* __restrict__ ridx, float* __restrict__ out) {
  size_t tid = (size_t)blockIdx.x * 256 + threadIdx.x;
  const size_t total = (size_t)BATCH*NS*NC;
  if (tid >= total) return;
  int n = (int)(tid % NC);
  size_t row = tid / NC;
  int b = (int)(row / NS);
  int i = (int)(row % NS);
  int idx = ridx[(size_t)b*NS + i];
  out[OUT_SSEG + tid] = out[OUT_SEG + ((size_t)b*HW + idx)*NC + n];
}

extern "C" void kernel_launch(void* const* d_in, const int* in_sizes, int n_in,
                              void* d_out, int out_size, void* d_ws, size_t ws_size,
                              hipStream_t stream) {
  (void)in_sizes; (void)n_in; (void)out_size; (void)ws_size;
  const float* logit     = (const float*)d_in[0];
  const float* embedding = (const float*)d_in[1];
  const float* param     = (const float*)d_in[2];
  const int*   gt_seg    = (const int*)d_in[3];
  const int*   ridx      = (const int*)d_in[4];
  float* out = (float*)d_out;
  float* ws  = (float*)d_ws;

  gather_kernel<<<dim3(BATCH), 256, 0, stream>>>(logit, embedding, param, gt_seg, ridx, out, ws);
  meanshift_kernel<<<dim3(NSEED, BATCH), 32, 0, stream>>>(ws);
  nms_kernel<<<dim3(BATCH), 128, 0, stream>>>(ws, out);
  seg_kernel<<<dim3(48, BATCH), 256, 0, stream>>>(embedding, ws, out);
  const size_t total = (size_t)BATCH*NS*NC;
  sample_kernel<<<(unsigned)((total + 255) / 256), 256, 0, stream>>>(ridx, out);
}